// SSIMLoss_39324720562651
// MI455X (gfx1250) — compile-verified
//
#include <hip/hip_runtime.h>
#include <stdint.h>

// ---------------- problem constants ----------------
#define IMG_W   384
#define IMG_H   384
#define OUT_W   378            // 384 - 7 + 1
#define OUT_H   378
#define NBATCH  64
#define TR      27             // output rows per tile (378 = 27 * 14)
#define TILES   14
#define IR      (TR + 6)       // input rows per tile = 33
#define RB      392            // padded row buffer (384 + 8 halo pad)
#define NTHREADS 384

typedef __attribute__((ext_vector_type(2))) float v2f;
typedef __attribute__((ext_vector_type(8))) float v8f;

// ---- CDNA5 async global->LDS staging (ASYNCcnt path) ----
// Copies 16 bytes per active lane from global memory into LDS without a VGPR
// round trip. Low 32 bits of a flat pointer to __shared__ are the LDS offset.
__device__ __forceinline__ void async_copy_b128(uint32_t lds_off, const void* gptr) {
    uint64_t ga = (uint64_t)(uintptr_t)gptr;
    asm volatile("global_load_async_to_lds_b128 %0, %1, off"
                 :: "v"(lds_off), "v"(ga) : "memory");
}
__device__ __forceinline__ void wait_asynccnt0() {
    asm volatile("s_wait_asynccnt 0x0" ::: "memory");
}

// Issue one 384-float row of X and one of Y into LDS buffer p.
// Lanes 0..95 move X (16B each), lanes 96..191 move Y. Waves 6..11 idle here.
__device__ __forceinline__ void issue_row(const float* __restrict__ Xb,
                                          const float* __restrict__ Yb,
                                          int r, int p,
                                          float (*xrow)[RB], float (*yrow)[RB],
                                          int tid) {
    if (tid < 96) {
        const float* g = Xb + (size_t)r * IMG_W + tid * 4;
        uint32_t l = (uint32_t)(uintptr_t)&xrow[p][tid * 4];
        async_copy_b128(l, g);
    } else if (tid < 192) {
        int t = tid - 96;
        const float* g = Yb + (size_t)r * IMG_W + t * 4;
        uint32_t l = (uint32_t)(uintptr_t)&yrow[p][t * 4];
        async_copy_b128(l, g);
    }
}

// ---------------- main fused SSIM kernel ----------------
__global__ __launch_bounds__(NTHREADS, 1)
void ssim_main_kernel(const float* __restrict__ X, const float* __restrict__ Y,
                      const float* __restrict__ drange,
                      float* __restrict__ partial) {
    __shared__ __align__(16) float xrow[2][RB];   // double-buffered raw rows
    __shared__ __align__(16) float yrow[2][RB];
    __shared__ float ring[5][7][IMG_W];           // 7-row vertical ring, 5 quantities
    __shared__ float red[NTHREADS];

    const int tid  = threadIdx.x;
    const int wg   = blockIdx.x;
    const int b    = wg / TILES;
    const int tile = wg % TILES;
    const int r0   = tile * TR;

    // zero the vertical ring and the halo pads (pads are never overwritten)
    for (int i = tid; i < 5 * 7 * IMG_W; i += NTHREADS)
        (&ring[0][0][0])[i] = 0.0f;
    if (tid < RB - IMG_W) {
        xrow[0][IMG_W + tid] = 0.0f; xrow[1][IMG_W + tid] = 0.0f;
        yrow[0][IMG_W + tid] = 0.0f; yrow[1][IMG_W + tid] = 0.0f;
    }

    const float L    = drange[b];
    const float C1   = (0.01f * L) * (0.01f * L);
    const float C2   = (0.03f * L) * (0.03f * L);
    const float inv  = 1.0f / 49.0f;
    const float COVN = 49.0f / 48.0f;

    const float* Xb = X + (size_t)b * IMG_W * IMG_H;
    const float* Yb = Y + (size_t)b * IMG_W * IMG_H;

    float Vx = 0.f, Vy = 0.f, Vxx = 0.f, Vxy = 0.f, Vyy = 0.f;
    float acc = 0.f;

    // prime the pipeline: row r0 -> buffer 0 (DMA runs under the init barrier)
    issue_row(Xb, Yb, r0, 0, xrow, yrow, tid);
    __syncthreads();   // ring init + pads visible

    for (int rr = 0; rr < IR; ++rr) {
        const int p = rr & 1;
        wait_asynccnt0();        // my wave's async copies for buffer p done
        __syncthreads();         // everyone's copies done; prev buffer free
        if (rr + 1 < IR)
            issue_row(Xb, Yb, r0 + rr + 1, p ^ 1, xrow, yrow, tid);

        // horizontal 7-tap sums for x, y, x^2, xy, y^2 at column tid
        float hx = 0.f, hy = 0.f, hxx = 0.f, hxy = 0.f, hyy = 0.f;
        #pragma unroll
        for (int k = 0; k < 7; ++k) {
            float xv = xrow[p][tid + k];
            float yv = yrow[p][tid + k];
            hx += xv; hy += yv;
            hxx = fmaf(xv, xv, hxx);
            hxy = fmaf(xv, yv, hxy);
            hyy = fmaf(yv, yv, hyy);
        }

        // vertical sliding window via 7-deep ring (thread-private column)
        const int s = rr % 7;
        Vx  += hx  - ring[0][s][tid];
        Vy  += hy  - ring[1][s][tid];
        Vxx += hxx - ring[2][s][tid];
        Vxy += hxy - ring[3][s][tid];
        Vyy += hyy - ring[4][s][tid];
        ring[0][s][tid] = hx;  ring[1][s][tid] = hy;
        ring[2][s][tid] = hxx; ring[3][s][tid] = hxy; ring[4][s][tid] = hyy;

        if (rr >= 6 && tid < OUT_W) {
            float ux  = Vx  * inv, uy  = Vy  * inv;
            float uxx = Vxx * inv, uxy = Vxy * inv, uyy = Vyy * inv;
            float vx  = COVN * (uxx - ux * ux);
            float vy  = COVN * (uyy - uy * uy);
            float vxy = COVN * (uxy - ux * uy);
            float N1 = 2.0f * ux * uy + C1;
            float N2 = 2.0f * vxy + C2;
            float D1 = ux * ux + uy * uy + C1;
            float D2 = vx + vy + C2;
            acc += (N1 * N2) / (D1 * D2);
        }
    }

    // deterministic workgroup tree reduction
    red[tid] = acc;
    __syncthreads();
    for (int st = NTHREADS / 2; st > 0; st >>= 1) {
        if (tid < st) red[tid] += red[tid + st];
        __syncthreads();
    }
    if (tid == 0) partial[wg] = red[0];
}

// ---------------- WMMA-based final reduction (one wave32) ----------------
// A = ones(16x4): D[m][n] = sum_k B[k][n] + C[m][n]. Since A is all-ones, the
// total over row 0 of D equals the sum of ALL 64 packed B values, independent
// of B's lane layout. 14 iterations fold all 896 partials, fixed order.
__global__ __launch_bounds__(32, 1)
void ssim_reduce_kernel(const float* __restrict__ partials,
                        float* __restrict__ out, int n64, float scale) {
    const int lane = threadIdx.x;          // 0..31, full wave active (EXEC all 1s)
    v2f a; a.x = 1.0f; a.y = 1.0f;         // all-ones A matrix
    v8f c = {};
    for (int i = 0; i < n64; ++i) {
        v2f bv;
        bv.x = partials[i * 64 + lane];
        bv.y = partials[i * 64 + 32 + lane];
        c = __builtin_amdgcn_wmma_f32_16x16x4_f32(
                /*neg_a=*/false, a, /*neg_b=*/false, bv,
                /*c_mod=*/(short)0, c, /*reuse_a=*/false, /*reuse_b=*/false);
    }
    __shared__ float red[32];
    red[lane] = c[0];                      // VGPR0: M=0 row (lanes 0..15)
    __syncthreads();
    if (lane == 0) {
        float t = 0.f;
        for (int i = 0; i < 16; ++i) t += red[i];   // sum row M=0 over N
        out[0] = 1.0f - t * scale;
    }
}

// ---------------- launch ----------------
extern "C" void kernel_launch(void* const* d_in, const int* in_sizes, int n_in,
                              void* d_out, int out_size, void* d_ws, size_t ws_size,
                              hipStream_t stream) {
    const float* X  = (const float*)d_in[0];
    const float* Y  = (const float*)d_in[1];
    const float* dr = (const float*)d_in[2];
    float* out      = (float*)d_out;
    float* partial  = (float*)d_ws;        // 896 floats of scratch

    const int nwg = NBATCH * TILES;        // 896
    ssim_main_kernel<<<nwg, NTHREADS, 0, stream>>>(X, Y, dr, partial);

    const float scale = 1.0f / ((float)NBATCH * (float)OUT_W * (float)OUT_H);
    ssim_reduce_kernel<<<1, 32, 0, stream>>>(partial, out, nwg / 64, scale);
}